// TSApproximation_3676492005594
// MI455X (gfx1250) — compile-verified
//
#include <hip/hip_runtime.h>

// out[c,p] = bias[c] + sum_j coeff[c,j,0]*x[p,0]^j + coeff[c,j,1]*x[p,1]^j
//
// VALU-optimized packed-Horner kernel:
//  - the two DIMS live in the two halves of v_pk_fma_f32
//  - coefficient (dim0,dim1) pairs are uniform, contiguous 8B -> s_load'd
//    SGPR pairs consumed DIRECTLY by v_pk_fma_f32 (inline asm, "s" constraint)
//  - packed-F32 VOP3P requires OPSEL/OPSEL_HI == 0 (spelled explicitly)
//  - Horner chain STARTS as acc = k15*x + k14 with BOTH coefficients in SGPR
//    pairs (2 scalar operands) -- eliminates the 12 v_mov_b64 acc-inits
//  - bias folded into the j=0 coefficient via s_add_f32 (SALU float), and the
//    j=0 FMA also reads it from SGPRs -- eliminates 3 more v_mov_b64
//  - 4 pixels per thread: 12 independent Horner chains (ILP); 2x
//    global_load_b128 in, 3x global_store_b128 out, all coalesced
//  - global_prefetch_b8 32KB ahead of the streaming read (gfx1250 path)

typedef float v2f __attribute__((ext_vector_type(2)));

static constexpr int kNumTerms     = 16;
static constexpr int kNumPix       = 2048 * 2048;
static constexpr int kPixPerThread = 4;

// acc = acc * x + k, k in an even-aligned SGPR pair (1 constant-bus value)
__device__ __forceinline__ v2f pk_fma_s(v2f acc, v2f x, v2f k) {
    asm("v_pk_fma_f32 %0, %0, %1, %2 op_sel:[0,0,0] op_sel_hi:[0,0,0]"
        : "+v"(acc) : "v"(x), "s"(k));
    return acc;
}
// d = a * x + k with a and k BOTH in SGPR pairs (2 constant-bus values);
// used for the first Horner step so no v_mov_b64 acc-init is needed.
__device__ __forceinline__ v2f pk_fma_ss(v2f a, v2f x, v2f k) {
    v2f d;
    asm("v_pk_fma_f32 %0, %1, %2, %3 op_sel:[0,0,0] op_sel_hi:[0,0,0]"
        : "=v"(d) : "s"(a), "v"(x), "s"(k));
    return d;
}

__global__ __launch_bounds__(256) void ts_poly_kernel(
    const float* __restrict__ x,       // [P][2]
    const float* __restrict__ coeff,   // [3][16][2]
    const float* __restrict__ bias,    // [3]
    float* __restrict__ out)           // [3][P]
{
    const int tid = blockIdx.x * blockDim.x + threadIdx.x;   // 4 pixels / thread
    const size_t p0 = (size_t)tid * kPixPerThread;
    if (p0 >= (size_t)kNumPix) return;

    // Speculative streaming prefetch ~32KB ahead (global_prefetch_b8).
    __builtin_prefetch((const char*)x + p0 * 8 + 32768, 0, 1);

    // 4 pixels x 2 dims = 32B: two coalesced 16B loads.
    const float4* __restrict__ xv4 = reinterpret_cast<const float4*>(x);
    const float4 xAB = xv4[tid * 2 + 0];
    const float4 xCD = xv4[tid * 2 + 1];
    v2f xp[kPixPerThread];
    xp[0] = (v2f){xAB.x, xAB.y};
    xp[1] = (v2f){xAB.z, xAB.w};
    xp[2] = (v2f){xCD.x, xCD.y};
    xp[3] = (v2f){xCD.z, xCD.w};

    // Uniform coefficient pairs (stay in SGPRs).
    const v2f* __restrict__ cf = reinterpret_cast<const v2f*>(coeff);

    #pragma unroll
    for (int c = 0; c < 3; ++c) {
        // Fold bias into the constant term of the dim-0 polynomial (s_add_f32).
        v2f k0b = cf[c * kNumTerms + 0];
        k0b.x += bias[c];

        const v2f k15 = cf[c * kNumTerms + 15];
        const v2f k14 = cf[c * kNumTerms + 14];

        v2f acc[kPixPerThread];
        // First Horner step straight from SGPRs: acc = k15*x + k14.
        #pragma unroll
        for (int q = 0; q < kPixPerThread; ++q)
            acc[q] = pk_fma_ss(k15, xp[q], k14);

        #pragma unroll
        for (int j = kNumTerms - 3; j >= 1; --j) {
            const v2f k = cf[c * kNumTerms + j];
            #pragma unroll
            for (int q = 0; q < kPixPerThread; ++q)
                acc[q] = pk_fma_s(acc[q], xp[q], k);   // v_pk_fma_f32, k from SGPR pair
        }
        // j = 0 with bias folded in; k0b is uniform -> also an SGPR pair.
        #pragma unroll
        for (int q = 0; q < kPixPerThread; ++q)
            acc[q] = pk_fma_s(acc[q], xp[q], k0b);

        // out = dim0 poly + dim1 poly (bias already inside), 16B store per plane.
        float4 o;
        o.x = acc[0].x + acc[0].y;
        o.y = acc[1].x + acc[1].y;
        o.z = acc[2].x + acc[2].y;
        o.w = acc[3].x + acc[3].y;
        reinterpret_cast<float4*>(out + (size_t)c * kNumPix)[tid] = o;
    }
}

extern "C" void kernel_launch(void* const* d_in, const int* in_sizes, int n_in,
                              void* d_out, int out_size, void* d_ws, size_t ws_size,
                              hipStream_t stream) {
    const float* x     = (const float*)d_in[0];   // P*2 floats
    const float* coeff = (const float*)d_in[1];   // 3*16*2 floats
    const float* bias  = (const float*)d_in[2];   // 3 floats
    float* out = (float*)d_out;                   // 3*P floats

    const int threads  = 256;
    const int nthreads = kNumPix / kPixPerThread; // 4 pixels per thread
    const int blocks   = (nthreads + threads - 1) / threads;  // 4096
    ts_poly_kernel<<<blocks, threads, 0, stream>>>(x, coeff, bias, out);
}